// KernelAttn_56530359550029
// MI455X (gfx1250) — compile-verified
//
#include <hip/hip_runtime.h>

typedef __attribute__((ext_vector_type(2))) float v2f;
typedef __attribute__((ext_vector_type(8))) float v8f;

#define S_LEN 8192
#define DIM   64
#define BATCH 64

#define K1_WAVES  4   // waves per block, kernel 1
#define K1_CHUNKS 16  // blocks along S per batch, kernel 1
#define K2_WAVES  8   // waves per block, kernel 2

// ELU(x)+1 == x+1 for x>0, exp(x) for x<=0
__device__ __forceinline__ float feature_map(float x) {
    return x > 0.0f ? x + 1.0f : __expf(x);
}

// D = A(16x4 f32) * B(4x16 f32) + C(16x16 f32), wave32 WMMA
__device__ __forceinline__ v8f wmma4(v2f a, v2f b, v8f c) {
    return __builtin_amdgcn_wmma_f32_16x16x4_f32(false, a, false, b, (short)0, c,
                                                 false, false);
}

__global__ void zero_ws_kernel(float* __restrict__ p, int n) {
    int i = blockIdx.x * blockDim.x + threadIdx.x;
    if (i < n) p[i] = 0.0f;
}

// Kernel 1: KV[b] += phi(K)^T V over this wave's S-slice, KS[b] += sum phi(K)
__global__ void __launch_bounds__(K1_WAVES * 32)
linattn_kv_ksum(const float* __restrict__ K, const float* __restrict__ V,
                float* __restrict__ KV, float* __restrict__ KS) {
    const int b    = blockIdx.x;
    const int wave = threadIdx.x >> 5;
    const int lane = threadIdx.x & 31;
    const int half = lane >> 4;   // 0: K rows {0,1}; 1: K rows {2,3} of the 16x4 frag
    const int l16  = lane & 15;

    const int slice  = blockIdx.y * K1_WAVES + wave;          // 0..63
    const int sCount = S_LEN / (K1_CHUNKS * K1_WAVES);        // 128
    const int sBase  = slice * sCount;

    const float* Kb = K + (size_t)b * S_LEN * DIM;
    const float* Vb = V + (size_t)b * S_LEN * DIM;

    v8f acc[4][4];
#pragma unroll
    for (int i = 0; i < 4; ++i)
#pragma unroll
        for (int j = 0; j < 4; ++j)
#pragma unroll
            for (int r = 0; r < 8; ++r) acc[i][j][r] = 0.0f;

    float ks[4] = {0.f, 0.f, 0.f, 0.f};

    for (int s0 = sBase; s0 < sBase + sCount; s0 += 4) {
        const int srow = s0 + half * 2;
        v2f afr[4], bfr[4];
        // A = phi(K)^T tile: A[M=d][Ks=s_local]; lane holds M=l16, Ks=half*2+{0,1}
#pragma unroll
        for (int td = 0; td < 4; ++td) {
            float x0 = feature_map(Kb[(size_t)srow * DIM + td * 16 + l16]);
            float x1 = feature_map(Kb[(size_t)(srow + 1) * DIM + td * 16 + l16]);
            afr[td].x = x0;
            afr[td].y = x1;
            ks[td] += x0 + x1;   // fold ksum into the same loads
        }
        // B = V tile: B[Ks=s_local][N=e]; lane holds Ks=half*2+{0,1}, N=l16
#pragma unroll
        for (int te = 0; te < 4; ++te) {
            bfr[te].x = Vb[(size_t)srow * DIM + te * 16 + l16];
            bfr[te].y = Vb[(size_t)(srow + 1) * DIM + te * 16 + l16];
        }
#pragma unroll
        for (int td = 0; td < 4; ++td)
#pragma unroll
            for (int te = 0; te < 4; ++te)
                acc[td][te] = wmma4(afr[td], bfr[te], acc[td][te]);
    }

    // ksum: halves of the wave covered different s-subsets of the same d
#pragma unroll
    for (int td = 0; td < 4; ++td) {
        float tot = ks[td] + __shfl_xor(ks[td], 16, 32);
        if (half == 0)
            atomicAdd(&KS[b * DIM + td * 16 + l16], tot);
    }

    // kv partials: C/D layout -> VGPR r, lane: (M = td*16 + r + half*8, N = te*16 + l16)
#pragma unroll
    for (int td = 0; td < 4; ++td)
#pragma unroll
        for (int te = 0; te < 4; ++te)
#pragma unroll
            for (int r = 0; r < 8; ++r) {
                int m = td * 16 + r + half * 8;
                int n = te * 16 + l16;
                atomicAdd(&KV[((size_t)b * DIM + m) * DIM + n], acc[td][te][r]);
            }
}

// Kernel 2: O = (phi(Q) @ KV) / max(phi(Q) . KS, 1e-4), KV/KS cached in LDS
__global__ void __launch_bounds__(K2_WAVES * 32)
linattn_out(const float* __restrict__ Q, const float* __restrict__ KV,
            const float* __restrict__ KS, float* __restrict__ O) {
    __shared__ float sKV[DIM * DIM];
    __shared__ float sKS[DIM];

    const int b    = blockIdx.x;
    const int tid  = threadIdx.x;
    const int wave = tid >> 5;
    const int lane = tid & 31;
    const int half = lane >> 4;
    const int l16  = lane & 15;

    for (int i = tid; i < DIM * DIM; i += K2_WAVES * 32)
        sKV[i] = KV[(size_t)b * DIM * DIM + i];
    if (tid < DIM) sKS[tid] = KS[b * DIM + tid];
    __syncthreads();

    const int s0 = (blockIdx.y * K2_WAVES + wave) * 16;
    const float* Qrow = Q + ((size_t)b * S_LEN + s0 + l16) * DIM;

    v8f acc[4], dacc;
#pragma unroll
    for (int te = 0; te < 4; ++te)
#pragma unroll
        for (int r = 0; r < 8; ++r) acc[te][r] = 0.0f;
#pragma unroll
    for (int r = 0; r < 8; ++r) dacc[r] = 0.0f;

    for (int kk = 0; kk < DIM; kk += 4) {
        const int krow = kk + half * 2;
        // A = phi(Q) frag: lane holds M=l16 (s-row), Kd=half*2+{0,1}; contiguous 64-bit load
        v2f a;
        a.x = feature_map(Qrow[krow]);
        a.y = feature_map(Qrow[krow + 1]);
        // B = kv frags from LDS: lane holds Kd=half*2+{0,1}, N=l16
#pragma unroll
        for (int te = 0; te < 4; ++te) {
            v2f bf;
            bf.x = sKV[krow * DIM + te * 16 + l16];
            bf.y = sKV[(krow + 1) * DIM + te * 16 + l16];
            acc[te] = wmma4(a, bf, acc[te]);
        }
        // denominator tile: B columns all equal ksum -> row-matched deno per C-row
        v2f bd;
        bd.x = sKS[krow];
        bd.y = sKS[krow + 1];
        dacc = wmma4(a, bd, dacc);
    }

    float* Ob = O + ((size_t)b * S_LEN + s0) * DIM;
#pragma unroll
    for (int r = 0; r < 8; ++r) {
        const int m = r + half * 8;
        const float inv = 1.0f / fmaxf(dacc[r], 1e-4f);
#pragma unroll
        for (int te = 0; te < 4; ++te)
            Ob[(size_t)m * DIM + te * 16 + l16] = acc[te][r] * inv;
    }
}

extern "C" void kernel_launch(void* const* d_in, const int* in_sizes, int n_in,
                              void* d_out, int out_size, void* d_ws, size_t ws_size,
                              hipStream_t stream) {
    const float* q = (const float*)d_in[0];
    const float* k = (const float*)d_in[1];
    const float* v = (const float*)d_in[2];
    float* out = (float*)d_out;

    float* KV = (float*)d_ws;                       // [B][64][64]
    float* KS = KV + (size_t)BATCH * DIM * DIM;     // [B][64]
    const int nzero = BATCH * DIM * DIM + BATCH * DIM;

    zero_ws_kernel<<<(nzero + 255) / 256, 256, 0, stream>>>(KV, nzero);
    linattn_kv_ksum<<<dim3(BATCH, K1_CHUNKS), K1_WAVES * 32, 0, stream>>>(k, v, KV, KS);
    linattn_out<<<dim3(BATCH, S_LEN / (16 * K2_WAVES)), K2_WAVES * 32, 0, stream>>>(q, KV, KS, out);
}